// VPLayer_21234318311937
// MI455X (gfx1250) — compile-verified
//
#include <hip/hip_runtime.h>
#include <math.h>

#define BB      512
#define TT      16385
#define NN      16384
#define THREADS 256
#define PER     64          // NN / THREADS

// output layout (flat f32): coeffs[512*2] | x_hat[512*16384] | res[512*16384] | r2[512]
#define O_XH   1024
#define O_RES  (1024 + (size_t)BB * NN)
#define O_R2   (1024 + (size_t)2 * BB * NN)

typedef float        v2f __attribute__((ext_vector_type(2)));
typedef float        v8f __attribute__((ext_vector_type(8)));
typedef unsigned int v4u __attribute__((ext_vector_type(4)));
typedef int          v4i __attribute__((ext_vector_type(4)));
typedef int          v8i __attribute__((ext_vector_type(8)));

// padded LDS index: one pad dword per 64 elements -> conflict-free serial walks
__device__ __forceinline__ int pidx(int n) { return n + (n >> 6); }
#define XS_SIZE (TT + (TT >> 6) + 4)   // 16645

#if defined(__gfx1250__) && __has_builtin(__builtin_amdgcn_tensor_load_to_lds) && \
    __has_builtin(__builtin_amdgcn_s_wait_tensorcnt)
#define USE_TDM 1
#endif

__global__ __launch_bounds__(THREADS) void vp_kernel(
    const float* __restrict__ x, const float* __restrict__ params,
    float* __restrict__ out)
{
    __shared__ float  xs[XS_SIZE];        // staged x row (TDM-padded layout)
    __shared__ float  ssum[THREADS];      // per-thread trapezoid partial sums
    __shared__ float  sP[256];            // 16x16 within-chunk inclusive prefixes
    __shared__ float  sOff[16];           // exclusive chunk offsets
    __shared__ double red[THREADS][5];    // f64 reduction scratch
    __shared__ float  sc[2];              // broadcast c0, c1

    const int b = blockIdx.x;
    const int t = threadIdx.x;
    const float r  = params[0];
    const float ny = params[1];
    const float* __restrict__ xrow = x + (size_t)b * TT;

    // ---- stage row into LDS with padded layout (pad 1 dword per 64) ----
#if defined(USE_TDM)
    if (t == 0) {
        // D# group 0: count=1 | lds_addr | global_addr[56:0] | type=2
        const unsigned int lds_addr = (unsigned int)(uintptr_t)(void*)&xs[0];
        const unsigned long long ga = (unsigned long long)(uintptr_t)xrow;
        v4u g0;
        g0[0] = 1u;                                   // count=1, user mode, no gather
        g0[1] = lds_addr;                             // lds_addr (bytes)
        g0[2] = (unsigned int)(ga & 0xFFFFFFFFu);     // global_addr[31:0]
        g0[3] = (unsigned int)((ga >> 32) & 0x1FFFFFFu) | (2u << 30);  // ga[56:32] | type=2
        // D# group 1: data_size=4B, pad_enable, pad_interval=64dw, pad_amount=1dw,
        //             tensor_dim0 = tile_dim0 = 16385, tensor_dim1 = 1, stride0 = 16385
        v8i g1;
        g1[0] = (2 << 16) | (1 << 20) | (5 << 22);    // data_size|pad_enable|pad_interval
        g1[1] = (int)((TT & 0xFFFF) << 16);           // tensor_dim0[15:0] << 16
        g1[2] = (int)((1u << 16) | (TT >> 16));       // tensor_dim1[15:0]<<16 | tensor_dim0[31:16]
        g1[3] = (int)((TT & 0xFFFF) << 16);           // tile_dim0[15:0] << 16
        g1[4] = 0;                                    // tile_dim1=0 (unused), tile_dim2=0
        g1[5] = TT;                                   // tensor_dim0_stride[31:0]
        g1[6] = 0;
        g1[7] = 0;
        v4i g2 = {};                                  // dims 2/3 unused (zero)
        v4i g3 = {};
#if __clang_major__ >= 23
        v8i g4 = {};
        __builtin_amdgcn_tensor_load_to_lds(g0, g1, g2, g3, g4, 0);
#else
        __builtin_amdgcn_tensor_load_to_lds(g0, g1, g2, g3, 0);
#endif
        __builtin_amdgcn_s_wait_tensorcnt((short)0);
    }
    __syncthreads();
#else
    for (int k = 0; k < PER; ++k) {
        int n = t + k * THREADS;                 // 0..16383
        xs[pidx(n)] = xrow[n];
    }
    if (t == 0) xs[pidx(NN)] = xrow[NN];
    __syncthreads();
#endif

    // ---- pass A: per-thread serial trapezoid partial sum over 64 elems ----
    const int base = t * PER;
    {
        float x0  = xs[pidx(base)];
        float run = 0.f;
        for (int i = 0; i < PER; ++i) {
            float x1 = xs[pidx(base + i + 1)];
            run += 0.5f * (x0 + x1);
            x0 = x1;
        }
        ssum[t] = run;
    }
    __syncthreads();

    // ---- block scan of 256 partials ----
#if defined(__gfx1250__) && __has_builtin(__builtin_amdgcn_wmma_f32_16x16x4_f32)
    // D = S^T x L^T : D[c][i] = sum_{k<=i} s[16c+k]  (16 parallel 16-elt scans)
    if (t < 32) {
        const int m = t & 15;     // A row (chunk) == B column for this lane
        const int h = t >> 4;
        v8f acc = {};
        #pragma unroll
        for (int j = 0; j < 4; ++j) {
            const int k0 = 4 * j + 2 * h;
            v2f a, bm;
            a.x  = ssum[16 * m + k0];
            a.y  = ssum[16 * m + k0 + 1];
            bm.x = (k0     <= m) ? 1.f : 0.f;   // upper-triangular ones (incl diag)
            bm.y = (k0 + 1 <= m) ? 1.f : 0.f;
            acc = __builtin_amdgcn_wmma_f32_16x16x4_f32(
                      false, a, false, bm, (short)0, acc, false, false);
        }
        #pragma unroll
        for (int p = 0; p < 8; ++p)
            sP[(p + 8 * h) * 16 + m] = acc[p];
    }
    __syncthreads();
    if (t == 0) {                         // exclusive scan of 16 chunk totals
        float o = 0.f;
        for (int c = 0; c < 16; ++c) { sOff[c] = o; o += sP[c * 16 + 15]; }
    }
    __syncthreads();
    const int   cch = t >> 4, pos = t & 15;
    const float E   = sOff[cch] + (pos ? sP[cch * 16 + (pos - 1)] : 0.f);
#else
    if (t == 0) {                         // scalar fallback: exclusive scan
        float o = 0.f;
        for (int n = 0; n < THREADS; ++n) { sP[n] = o; o += ssum[n]; }
    }
    __syncthreads();
    const float E = sP[t];
#endif

    // ---- pass B: recompute u,v; accumulate Gram + projection in f64 ----
    double su2 = 0, suv = 0, sv2 = 0, sxu = 0, sxv = 0;
    {
        float x0  = xs[pidx(base)];
        float run = E;
        for (int i = 0; i < PER; ++i) {
            float x1 = xs[pidx(base + i + 1)];
            run += 0.5f * (x0 + x1);
            float u = ny + run;
            float v = powf(u, r);
            double du = u, dv = v, dx = x1;
            su2 += du * du;  suv += du * dv;  sv2 += dv * dv;
            sxu += dx * du;  sxv += dx * dv;
            x0 = x1;
        }
    }
    red[t][0] = su2; red[t][1] = suv; red[t][2] = sv2;
    red[t][3] = sxu; red[t][4] = sxv;
    __syncthreads();
    for (int off = THREADS / 2; off > 0; off >>= 1) {
        if (t < off)
            for (int j = 0; j < 5; ++j) red[t][j] += red[t + off][j];
        __syncthreads();
    }
    if (t == 0) {
        double G00 = red[0][0], G01 = red[0][1], G11 = red[0][2];
        double p0  = red[0][3], p1  = red[0][4];
        double det = G00 * G11 - G01 * G01;
        double c0  = (p0 * G11 - p1 * G01) / det;
        double c1  = (p1 * G00 - p0 * G01) / det;
        out[(size_t)b * 2 + 0] = (float)c0;
        out[(size_t)b * 2 + 1] = (float)c1;
        sc[0] = (float)c0;  sc[1] = (float)c1;
    }
    __syncthreads();

    // ---- pass C: x_hat, res (b128 stores), r2 in f64 ----
    const float c0 = sc[0], c1 = sc[1];
    float* __restrict__ oxh  = out + O_XH  + (size_t)b * NN;
    float* __restrict__ ores = out + O_RES + (size_t)b * NN;
    double sr2 = 0;
    {
        float x0  = xs[pidx(base)];
        float run = E;
        for (int i4 = 0; i4 < PER / 4; ++i4) {
            float4 vh, vr;
            float* ph = (float*)&vh;
            float* pr = (float*)&vr;
            #pragma unroll
            for (int q = 0; q < 4; ++q) {
                int   i  = i4 * 4 + q;
                float x1 = xs[pidx(base + i + 1)];
                run += 0.5f * (x0 + x1);
                float u  = ny + run;
                float v  = powf(u, r);
                float xh = c0 * u + c1 * v;
                float rs = x1 - xh;
                ph[q] = xh;  pr[q] = rs;
                sr2 += (double)rs * rs;
                x0 = x1;
            }
            *(float4*)(oxh  + base + i4 * 4) = vh;
            *(float4*)(ores + base + i4 * 4) = vr;
        }
    }
    red[t][0] = sr2;
    __syncthreads();
    for (int off = THREADS / 2; off > 0; off >>= 1) {
        if (t < off) red[t][0] += red[t + off][0];
        __syncthreads();
    }
    if (t == 0) out[O_R2 + b] = (float)red[0][0];
}

extern "C" void kernel_launch(void* const* d_in, const int* in_sizes, int n_in,
                              void* d_out, int out_size, void* d_ws, size_t ws_size,
                              hipStream_t stream)
{
    const float* x      = (const float*)d_in[0];
    const float* params = (const float*)d_in[1];
    float*       out    = (float*)d_out;
    vp_kernel<<<BB, THREADS, 0, stream>>>(x, params, out);
}